// GCNcleaner_541165879372
// MI455X (gfx1250) — compile-verified
//
#include <hip/hip_runtime.h>
#include <hip/hip_bf16.h>

#define N_NODES 100000
#define N_EDGES 3200000
#define IN_DIM  512
#define HID_DIM 16
#define N_TILES (N_NODES / 16)   // 6250 exact

typedef float v2f __attribute__((ext_vector_type(2)));
typedef float v8f __attribute__((ext_vector_type(8)));

// ---------------------------------------------------------------------------
// Kernel 1: h1 = X @ W1 via V_WMMA_F32_16X16X4_F32.
// One wave (32 lanes) computes one 16x16 output tile; 8 waves / 256-thread block.
// W1 (512x16 fp32 = 32 KB) staged in LDS once per block.
// ---------------------------------------------------------------------------
__global__ __launch_bounds__(256) void gcn_gemm_xw1(
    const float* __restrict__ X, const float* __restrict__ W1,
    float* __restrict__ h1) {
  __shared__ float sW[IN_DIM * HID_DIM];
  for (int i = threadIdx.x; i < IN_DIM * HID_DIM; i += 256) sW[i] = W1[i];
  __syncthreads();

  const int wave = threadIdx.x >> 5;
  const int lane = threadIdx.x & 31;
  const int tile = blockIdx.x * 8 + wave;       // wave-uniform -> EXEC stays full
  if (tile >= N_TILES) return;

  const int m  = lane & 15;   // row within tile for A; column for B/C
  const int hi = lane >> 4;   // K sub-pair selector (ISA A/B f32 layout)

  const float* __restrict__ xrow = X + (size_t)(tile * 16 + m) * IN_DIM;

  v8f c = {0.f, 0.f, 0.f, 0.f, 0.f, 0.f, 0.f, 0.f};

  #pragma unroll 4
  for (int k0 = 0; k0 < IN_DIM; k0 += 4) {
    const int k = k0 + 2 * hi;
    // A-frag 16x4: lanes 0-15 K={k0,k0+1}, lanes 16-31 K={k0+2,k0+3}
    v2f a;
    a.x = xrow[k];
    a.y = xrow[k + 1];
    // B-frag 4x16 from LDS, symmetric K striping, N = lane%16
    v2f b;
    b.x = sW[k * HID_DIM + m];
    b.y = sW[(k + 1) * HID_DIM + m];
    // (neg_a, A, neg_b, B, c_mod, C, reuse_a, reuse_b)
    c = __builtin_amdgcn_wmma_f32_16x16x4_f32(
        false, a, false, b, (short)0, c, false, false);
  }

  // C layout: VGPR v -> lanes 0-15: M=v, N=lane; lanes 16-31: M=v+8, N=lane-16
  float* __restrict__ out = h1 + (size_t)tile * 16 * HID_DIM;
  const int mbase = hi * 8;
  #pragma unroll
  for (int v = 0; v < 8; ++v)
    out[(mbase + v) * HID_DIM + m] = c[v];
}

// ---------------------------------------------------------------------------
// Kernel 2: init h2 = 0 and out = b2 (scatter targets, both L2-resident)
// ---------------------------------------------------------------------------
__global__ void gcn_init(float* __restrict__ h2, float* __restrict__ out,
                         const float* __restrict__ b2) {
  int i = blockIdx.x * blockDim.x + threadIdx.x;
  if (i < N_NODES * HID_DIM) h2[i] = 0.0f;
  if (i < N_NODES) out[i] = b2[0];
}

// ---------------------------------------------------------------------------
// Kernel 3: edge-parallel SpMM layer 1. 16 lanes per edge; coalesced gather of
// the h1 row, native fp32 atomic scatter into h2 (L2).
// ---------------------------------------------------------------------------
__global__ __launch_bounds__(256) void gcn_spmm1(
    const int* __restrict__ esrc, const int* __restrict__ edst,
    const float* __restrict__ ew, const float* __restrict__ h1,
    float* __restrict__ h2) {
  size_t gid = (size_t)blockIdx.x * 256 + threadIdx.x;
  size_t e = gid >> 4;
  if (e >= N_EDGES) return;
  int d = (int)(gid & 15);
  int s = esrc[e];
  int t = edst[e];
  float v = ew[e] * h1[(size_t)s * HID_DIM + d];
  __hip_atomic_fetch_add(&h2[(size_t)t * HID_DIM + d], v,
                         __ATOMIC_RELAXED, __HIP_MEMORY_SCOPE_AGENT);
}

// ---------------------------------------------------------------------------
// Kernel 4: z[i] = sum_d relu(h2[i,d] + b1[d]) * W2[d]   (fused bias/ReLU/GEMV)
// ---------------------------------------------------------------------------
__global__ void gcn_layer2(const float* __restrict__ h2,
                           const float* __restrict__ b1,
                           const float* __restrict__ W2,
                           float* __restrict__ z) {
  int i = blockIdx.x * blockDim.x + threadIdx.x;
  if (i >= N_NODES) return;
  const float* __restrict__ row = h2 + (size_t)i * HID_DIM;
  float acc = 0.0f;
  #pragma unroll
  for (int d = 0; d < HID_DIM; ++d) {
    float hv = row[d] + b1[d];
    hv = fmaxf(hv, 0.0f);
    acc = fmaf(hv, W2[d], acc);
  }
  z[i] = acc;
}

// ---------------------------------------------------------------------------
// Kernel 5: SpMM layer 2 (scalar): out[dst] += w * z[src]; out pre-seeded b2.
// ---------------------------------------------------------------------------
__global__ __launch_bounds__(256) void gcn_spmm2(
    const int* __restrict__ esrc, const int* __restrict__ edst,
    const float* __restrict__ ew, const float* __restrict__ z,
    float* __restrict__ out) {
  size_t e = (size_t)blockIdx.x * 256 + threadIdx.x;
  if (e >= N_EDGES) return;
  float v = ew[e] * z[esrc[e]];
  __hip_atomic_fetch_add(&out[edst[e]], v,
                         __ATOMIC_RELAXED, __HIP_MEMORY_SCOPE_AGENT);
}

extern "C" void kernel_launch(void* const* d_in, const int* in_sizes, int n_in,
                              void* d_out, int out_size, void* d_ws, size_t ws_size,
                              hipStream_t stream) {
  const float* x    = (const float*)d_in[0];
  const int*   esrc = (const int*)  d_in[1];
  const int*   edst = (const int*)  d_in[2];
  const float* ew   = (const float*)d_in[3];
  const float* W1   = (const float*)d_in[4];
  const float* b1   = (const float*)d_in[5];
  const float* W2   = (const float*)d_in[6];
  const float* b2   = (const float*)d_in[7];
  float* out = (float*)d_out;

  // workspace layout: h1 [N*16] | h2 [N*16] | z [N]
  float* h1 = (float*)d_ws;
  float* h2 = h1 + (size_t)N_NODES * HID_DIM;
  float* z  = h2 + (size_t)N_NODES * HID_DIM;

  // 1) dense GEMM with WMMA (6250 tiles, 8 waves/block)
  {
    int blocks = (N_TILES + 7) / 8;
    gcn_gemm_xw1<<<blocks, 256, 0, stream>>>(x, W1, h1);
  }
  // 2) init scatter targets
  {
    int total = N_NODES * HID_DIM;
    gcn_init<<<(total + 255) / 256, 256, 0, stream>>>(h2, out, b2);
  }
  // 3) SpMM layer 1 (16 lanes per edge)
  {
    size_t total = (size_t)N_EDGES * HID_DIM;
    int blocks = (int)((total + 255) / 256);
    gcn_spmm1<<<blocks, 256, 0, stream>>>(esrc, edst, ew, h1, h2);
  }
  // 4) bias + ReLU + W2 dot
  gcn_layer2<<<(N_NODES + 255) / 256, 256, 0, stream>>>(h2, b1, W2, z);
  // 5) SpMM layer 2
  gcn_spmm2<<<(N_EDGES + 255) / 256, 256, 0, stream>>>(esrc, edst, ew, z, out);
}